// EdgeGNN_14267881357876
// MI455X (gfx1250) — compile-verified
//
#include <hip/hip_runtime.h>

#define N_NODES 50000
#define N_EDGES 800000
#define NODE_IN 128
#define EDGE_IN 64
#define HIDDEN  128
#define OUT     2

typedef __attribute__((ext_vector_type(2))) float v2f;
typedef __attribute__((ext_vector_type(8))) float v8f;

// D = A(16x4 f32) * B(4x16 f32) + C(16x16 f32), exact fp32 via CDNA5 WMMA.
static __device__ __forceinline__ v8f wmma_f32_16x16x4(v2f a, v2f b, v8f c) {
  return __builtin_amdgcn_wmma_f32_16x16x4_f32(
      /*neg_a=*/false, a, /*neg_b=*/false, b,
      /*c_mod=*/(short)0, c, /*reuse_a=*/false, /*reuse_b=*/false);
}

// ---------------- zero scratch (agg + cnt) ----------------
__global__ void k_zero(float* __restrict__ buf, int n) {
  int t = blockIdx.x * blockDim.x + threadIdx.x;
  if (t < n) buf[t] = 0.0f;
}

// ---------------- fold Wl@Wc, Wr@Wc, bl@Wc (tiny) ----------------
__global__ void k_fold(const float* __restrict__ Wl, const float* __restrict__ Wr,
                       const float* __restrict__ bl, const float* __restrict__ Wc,
                       float* __restrict__ Wlc, float* __restrict__ Wrc,
                       float* __restrict__ blc) {
  int t = threadIdx.x;
  if (t < 256) {
    int k = t >> 1, o = t & 1;
    float s = 0.0f;
    for (int j = 0; j < HIDDEN; ++j) s += Wl[k * HIDDEN + j] * Wc[j * OUT + o];
    Wlc[k * OUT + o] = s;
  } else if (t < 512) {
    int tt = t - 256, k = tt >> 1, o = tt & 1;
    float s = 0.0f;
    for (int j = 0; j < HIDDEN; ++j) s += Wr[k * HIDDEN + j] * Wc[j * OUT + o];
    Wrc[k * OUT + o] = s;
  } else if (t < 514) {
    int o = t - 512;
    float s = 0.0f;
    for (int j = 0; j < HIDDEN; ++j) s += bl[j] * Wc[j * OUT + o];
    blc[o] = s;
  }
}

// ---------------- h = relu(x @ Wn + bn), WMMA fp32 ----------------
// block = 256 (8 waves). blockIdx.x -> 16-row tile, wave -> 16-col tile.
__global__ __launch_bounds__(256) void k_node_mlp(const float* __restrict__ x,
                                                  const float* __restrict__ Wn,
                                                  const float* __restrict__ bn,
                                                  float* __restrict__ h) {
  const int lane = threadIdx.x & 31;
  const int wave = threadIdx.x >> 5;
  const int lo = lane & 15, hi = lane >> 4;
  const int m0 = blockIdx.x * 16;
  const int n0 = wave * 16;
  const int rowA = m0 + lo;

  v8f acc = {};
  for (int k0 = 0; k0 < NODE_IN; k0 += 4) {
    // A 16x4: lane lo holds row M=lo; VGPR0/1 = K {k0+2*hi, k0+2*hi+1}
    v2f a = *(const v2f*)(x + (size_t)rowA * NODE_IN + k0 + 2 * hi);
    // B 4x16: VGPR0 = row K=k0 (low lanes) / k0+2 (high), VGPR1 = +1
    v2f b;
    b.x = Wn[(size_t)(k0 + 2 * hi)     * HIDDEN + n0 + lo];
    b.y = Wn[(size_t)(k0 + 2 * hi + 1) * HIDDEN + n0 + lo];
    acc = wmma_f32_16x16x4(a, b, acc);
  }
  const float bias = bn[n0 + lo];
#pragma unroll
  for (int r = 0; r < 8; ++r) {
    float v = acc[r] + bias;                       // C/D: M = r + 8*hi, N = lo
    h[(size_t)(m0 + r + 8 * hi) * HIDDEN + n0 + lo] = fmaxf(v, 0.0f);
  }
}

// ---------------- scatter: agg[dst] += h[src]; cnt[dst] += 1 ----------------
// 32 threads per edge, 4 features each (float4 read, 4 fp32 atomics).
__global__ void k_scatter(const float* __restrict__ h, const int* __restrict__ ei,
                          float* __restrict__ agg, float* __restrict__ cnt) {
  long long t = (long long)blockIdx.x * blockDim.x + threadIdx.x;
  int e = (int)(t >> 5);
  if (e >= N_EDGES) return;
  int g = ((int)t & 31) * 4;
  int s = ei[e];
  int d = ei[N_EDGES + e];
  float4 v = *(const float4*)(h + (size_t)s * HIDDEN + g);
  float* base = agg + (size_t)d * HIDDEN + g;
  unsafeAtomicAdd(base + 0, v.x);
  unsafeAtomicAdd(base + 1, v.y);
  unsafeAtomicAdd(base + 2, v.z);
  unsafeAtomicAdd(base + 3, v.w);
  if (((int)t & 31) == 0) unsafeAtomicAdd(cnt + d, 1.0f);
}

// ---------------- p = mean @ Wlc + h @ Wrc + blc   [N,2] ----------------
__global__ void k_node_out(const float* __restrict__ h, const float* __restrict__ agg,
                           const float* __restrict__ cnt, const float* __restrict__ Wlc,
                           const float* __restrict__ Wrc, const float* __restrict__ blc,
                           float* __restrict__ p) {
  int node = blockIdx.x * blockDim.x + threadIdx.x;
  if (node >= N_NODES) return;
  float inv = 1.0f / fmaxf(cnt[node], 1.0f);
  float p0 = blc[0], p1 = blc[1];
  const float* ar = agg + (size_t)node * HIDDEN;
  const float* hr = h + (size_t)node * HIDDEN;
  for (int k = 0; k < HIDDEN; k += 4) {
    float4 a  = *(const float4*)(ar + k);
    float4 hh = *(const float4*)(hr + k);
    float m0v = a.x * inv, m1v = a.y * inv, m2v = a.z * inv, m3v = a.w * inv;
    p0 += m0v * Wlc[(k + 0) * OUT + 0] + hh.x * Wrc[(k + 0) * OUT + 0]
        + m1v * Wlc[(k + 1) * OUT + 0] + hh.y * Wrc[(k + 1) * OUT + 0]
        + m2v * Wlc[(k + 2) * OUT + 0] + hh.z * Wrc[(k + 2) * OUT + 0]
        + m3v * Wlc[(k + 3) * OUT + 0] + hh.w * Wrc[(k + 3) * OUT + 0];
    p1 += m0v * Wlc[(k + 0) * OUT + 1] + hh.x * Wrc[(k + 0) * OUT + 1]
        + m1v * Wlc[(k + 1) * OUT + 1] + hh.y * Wrc[(k + 1) * OUT + 1]
        + m2v * Wlc[(k + 2) * OUT + 1] + hh.z * Wrc[(k + 2) * OUT + 1]
        + m3v * Wlc[(k + 3) * OUT + 1] + hh.w * Wrc[(k + 3) * OUT + 1];
  }
  p[(size_t)node * OUT + 0] = p0;
  p[(size_t)node * OUT + 1] = p1;
}

// ---------------- fused edge MLP + q-dot + gather + output ----------------
// block = 128 (4 waves); each wave owns one 16-edge tile.
// Per wave: 8 col-tiles of WMMA build e-tile (16x128, post-ReLU) in LDS,
// then lane (lo,hi) computes q[m=lo][o=hi] and writes the final logit.
__global__ __launch_bounds__(128) void k_edge_out(
    const float* __restrict__ ea, const float* __restrict__ We,
    const float* __restrict__ be, const float* __restrict__ Wc,
    const float* __restrict__ bc, const int* __restrict__ ei,
    const float* __restrict__ p, float* __restrict__ out) {
  __shared__ float etile[4][16][129];   // 129-float rows: bank = (m+k)%64, conflict-free
  const int lane = threadIdx.x & 31;
  const int wave = threadIdx.x >> 5;
  const int lo = lane & 15, hi = lane >> 4;
  const int tile = blockIdx.x * 4 + wave;
  const int e0 = tile * 16;

  for (int n0 = 0; n0 < HIDDEN; n0 += 16) {
    v8f acc = {};
    for (int k0 = 0; k0 < EDGE_IN; k0 += 4) {
      v2f a = *(const v2f*)(ea + (size_t)(e0 + lo) * EDGE_IN + k0 + 2 * hi);
      v2f b;
      b.x = We[(size_t)(k0 + 2 * hi)     * HIDDEN + n0 + lo];
      b.y = We[(size_t)(k0 + 2 * hi + 1) * HIDDEN + n0 + lo];
      acc = wmma_f32_16x16x4(a, b, acc);
    }
    float bias = be[n0 + lo];
#pragma unroll
    for (int r = 0; r < 8; ++r)
      etile[wave][r + 8 * hi][n0 + lo] = fmaxf(acc[r] + bias, 0.0f);
  }
  __syncthreads();

  float qv = 0.0f;
  for (int k = 0; k < HIDDEN; ++k)
    qv += etile[wave][lo][k] * Wc[k * OUT + hi];

  int e = e0 + lo;
  int s = ei[e];
  int d = ei[N_EDGES + e];
  float val = (p[(size_t)s * OUT + hi] + p[(size_t)d * OUT + hi] + qv) * (1.0f / 3.0f)
            + bc[hi];
  out[(size_t)e * OUT + hi] = val;
}

extern "C" void kernel_launch(void* const* d_in, const int* in_sizes, int n_in,
                              void* d_out, int out_size, void* d_ws, size_t ws_size,
                              hipStream_t stream) {
  const float* x  = (const float*)d_in[0];
  const int*   ei = (const int*)d_in[1];
  const float* ea = (const float*)d_in[2];
  const float* Wn = (const float*)d_in[3];
  const float* bn = (const float*)d_in[4];
  const float* We = (const float*)d_in[5];
  const float* be = (const float*)d_in[6];
  const float* Wl = (const float*)d_in[7];
  const float* bl = (const float*)d_in[8];
  const float* Wr = (const float*)d_in[9];
  const float* Wc = (const float*)d_in[10];
  const float* bc = (const float*)d_in[11];
  float* out = (float*)d_out;

  // Workspace layout (floats): h | agg | cnt | Wlc | Wrc | blc | p  (~52 MB)
  float* ws  = (float*)d_ws;
  float* h   = ws;                                   // N*H
  float* agg = h   + (size_t)N_NODES * HIDDEN;       // N*H (16B-aligned offset)
  float* cnt = agg + (size_t)N_NODES * HIDDEN;       // N
  float* Wlc = cnt + N_NODES;                        // H*OUT
  float* Wrc = Wlc + HIDDEN * OUT;                   // H*OUT
  float* blc = Wrc + HIDDEN * OUT;                   // OUT
  float* p   = blc + OUT;                            // N*OUT

  // 1) zero agg+cnt (contiguous) — required every call (graph replays)
  int zn = N_NODES * HIDDEN + N_NODES;
  k_zero<<<(zn + 255) / 256, 256, 0, stream>>>(agg, zn);
  // 2) fold the linear tail through Wc
  k_fold<<<1, 544, 0, stream>>>(Wl, Wr, bl, Wc, Wlc, Wrc, blc);
  // 3) h = relu(x @ Wn + bn)        (3125 row tiles x 8 col tiles)
  k_node_mlp<<<N_NODES / 16, 256, 0, stream>>>(x, Wn, bn, h);
  // 4) mean-aggregation scatter
  k_scatter<<<(N_EDGES * 32) / 256, 256, 0, stream>>>(h, ei, agg, cnt);
  // 5) p = mean@Wlc + h@Wrc + blc
  k_node_out<<<(N_NODES + 255) / 256, 256, 0, stream>>>(h, agg, cnt, Wlc, Wrc, blc, p);
  // 6) fused edge MLP + output      (50000 tiles / 4 per block)
  k_edge_out<<<N_EDGES / 64, 128, 0, stream>>>(ea, We, be, Wc, bc, ei, p, out);
}